// RNetwork_11433202942183
// MI455X (gfx1250) — compile-verified
//
#include <hip/hip_runtime.h>

#define NN 80000      // nodes
#define NE 1280000    // edges
#define ND 32         // node dim
#define ED 16         // edge dim
#define HID 64        // hidden
#define NG 256        // graphs

// LDS weight row stride (in floats) per K-pair row: 128 payload + 32 pad so the
// hf=1 half-wave maps to banks 32..63 (stride mod 64 == 32) -> conflict-free b64.
#define WSTRIDE 160

typedef __attribute__((ext_vector_type(2))) float v2f;
typedef __attribute__((ext_vector_type(8))) float v8f;

__device__ __forceinline__ void atomAddF(float* p, float v) {
  __hip_atomic_fetch_add(p, v, __ATOMIC_RELAXED, __HIP_MEMORY_SCOPE_AGENT);
}

__global__ __launch_bounds__(256) void zero_f(float* __restrict__ p, int n) {
  int i = blockIdx.x * 256 + threadIdx.x;
  if (i < n) p[i] = 0.f;
}

__global__ __launch_bounds__(256) void init_out_k(float* __restrict__ out,
                                                  const float* __restrict__ Wb) {
  if (threadIdx.x < NG) out[threadIdx.x] = Wb[0];
}

// Stage W [KIN,64] into LDS as K-pair-interleaved, padded rows:
//   wl[j*WSTRIDE + 2n + t] = W[(2j+t)*64 + n],  j = 0..KIN/2-1
// Lane fragment for k-step k, half hf, col n is then one aligned ds_load_b64 at
//   wl + (2k+hf)*WSTRIDE + 2n.
template <int KIN>
__device__ __forceinline__ void stage_weights(float* wl, const float* __restrict__ W,
                                              const float* __restrict__ Bias) {
  for (int i = threadIdx.x; i < KIN * HID; i += 256) {
    const int j = i >> 7;        // i / 128
    const int rem = i & 127;
    const int n = rem >> 1;
    const int t = rem & 1;
    wl[j * WSTRIDE + 2 * n + t] = W[(2 * j + t) * HID + n];
  }
  if (threadIdx.x < HID) wl[(KIN / 2) * WSTRIDE + threadIdx.x] = Bias[threadIdx.x];
  __syncthreads();
}

// ---------------------------------------------------------------------------
// Edge message GEMM + ReLU + scatter-add:
//   m_e = relu([h[src[e]], Xe[e]] @ W + b);  agg[dst[e]] += m_e
// One wave per 16-edge tile, 4x v_wmma_f32_16x16x4_f32 per K-step.
// ---------------------------------------------------------------------------
template <int HIN>
__global__ __launch_bounds__(256) void msg_kernel(
    const float* __restrict__ h, const float* __restrict__ xe,
    const long long* __restrict__ src, const long long* __restrict__ dst,
    const float* __restrict__ W, const float* __restrict__ Bias,
    float* __restrict__ agg) {
  constexpr int KIN = HIN + ED;
  __shared__ float wl[(KIN / 2) * WSTRIDE + HID];
  stage_weights<KIN>(wl, W, Bias);

  const int lane = threadIdx.x & 31;
  const int m    = lane & 15;   // row within 16-edge tile / output col within 16
  const int hf   = lane >> 4;   // half-wave select
  const int tile  = blockIdx.x * 8 + (threadIdx.x >> 5);
  const int ebase = tile * 16;

  const long long s = src[ebase + m];
  const float* __restrict__ hrow = h + (size_t)s * HIN;
  const float* __restrict__ xrow = xe + (size_t)(ebase + m) * ED;

  v8f c0{}, c1{}, c2{}, c3{};
#pragma unroll
  for (int k = 0; k < KIN / 4; ++k) {
    const int kk = 4 * k + 2 * hf;
    // A fragment: X[m][kk], X[m][kk+1]  (concat boundary is 4-aligned -> no straddle)
    v2f a = (4 * k < HIN) ? *(const v2f*)(hrow + kk)
                          : *(const v2f*)(xrow + (kk - HIN));
    // B fragments: single conflict-free ds_load_b64 each
    const float* wj = wl + (2 * k + hf) * WSTRIDE + 2 * m;
    v2f b0 = *(const v2f*)(wj);
    v2f b1 = *(const v2f*)(wj + 32);
    v2f b2 = *(const v2f*)(wj + 64);
    v2f b3 = *(const v2f*)(wj + 96);
    c0 = __builtin_amdgcn_wmma_f32_16x16x4_f32(false, a, false, b0, (short)0, c0, false, false);
    c1 = __builtin_amdgcn_wmma_f32_16x16x4_f32(false, a, false, b1, (short)0, c1, false, false);
    c2 = __builtin_amdgcn_wmma_f32_16x16x4_f32(false, a, false, b2, (short)0, c2, false, false);
    c3 = __builtin_amdgcn_wmma_f32_16x16x4_f32(false, a, false, b3, (short)0, c3, false, false);
  }

  const float* bb = wl + (KIN / 2) * WSTRIDE;
  const float bi0 = bb[m], bi1 = bb[m + 16], bi2 = bb[m + 32], bi3 = bb[m + 48];
#pragma unroll
  for (int r = 0; r < 8; ++r) {                // D row = r + 8*hf, col = m (+16*nt)
    const long long d = dst[ebase + r + 8 * hf];
    float* arow = agg + (size_t)d * HID + m;
    atomAddF(arow,      fmaxf(c0[r] + bi0, 0.f));
    atomAddF(arow + 16, fmaxf(c1[r] + bi1, 0.f));
    atomAddF(arow + 32, fmaxf(c2[r] + bi2, 0.f));
    atomAddF(arow + 48, fmaxf(c3[r] + bi3, 0.f));
  }
}

// ---------------------------------------------------------------------------
// Node update GEMM + ReLU:  y = relu([h, agg] @ W + b)   (dense, no atomics)
// ---------------------------------------------------------------------------
template <int HIN>
__global__ __launch_bounds__(256) void upd_kernel(
    const float* __restrict__ h, const float* __restrict__ agg,
    const float* __restrict__ W, const float* __restrict__ Bias,
    float* __restrict__ y) {
  constexpr int KIN = HIN + HID;
  __shared__ float wl[(KIN / 2) * WSTRIDE + HID];
  stage_weights<KIN>(wl, W, Bias);

  const int lane = threadIdx.x & 31;
  const int m    = lane & 15;
  const int hf   = lane >> 4;
  const int tile  = blockIdx.x * 8 + (threadIdx.x >> 5);
  const int nbase = tile * 16;

  const float* __restrict__ hrow = h + (size_t)(nbase + m) * HIN;
  const float* __restrict__ arow = agg + (size_t)(nbase + m) * HID;

  v8f c0{}, c1{}, c2{}, c3{};
#pragma unroll
  for (int k = 0; k < KIN / 4; ++k) {
    const int kk = 4 * k + 2 * hf;
    v2f a = (4 * k < HIN) ? *(const v2f*)(hrow + kk)
                          : *(const v2f*)(arow + (kk - HIN));
    const float* wj = wl + (2 * k + hf) * WSTRIDE + 2 * m;
    v2f b0 = *(const v2f*)(wj);
    v2f b1 = *(const v2f*)(wj + 32);
    v2f b2 = *(const v2f*)(wj + 64);
    v2f b3 = *(const v2f*)(wj + 96);
    c0 = __builtin_amdgcn_wmma_f32_16x16x4_f32(false, a, false, b0, (short)0, c0, false, false);
    c1 = __builtin_amdgcn_wmma_f32_16x16x4_f32(false, a, false, b1, (short)0, c1, false, false);
    c2 = __builtin_amdgcn_wmma_f32_16x16x4_f32(false, a, false, b2, (short)0, c2, false, false);
    c3 = __builtin_amdgcn_wmma_f32_16x16x4_f32(false, a, false, b3, (short)0, c3, false, false);
  }

  const float* bb = wl + (KIN / 2) * WSTRIDE;
  const float bi0 = bb[m], bi1 = bb[m + 16], bi2 = bb[m + 32], bi3 = bb[m + 48];
#pragma unroll
  for (int r = 0; r < 8; ++r) {
    float* yr = y + (size_t)(nbase + r + 8 * hf) * HID + m;
    yr[0]  = fmaxf(c0[r] + bi0, 0.f);
    yr[16] = fmaxf(c1[r] + bi1, 0.f);
    yr[32] = fmaxf(c2[r] + bi2, 0.f);
    yr[48] = fmaxf(c3[r] + bi3, 0.f);
  }
}

// ---------------------------------------------------------------------------
// Graph pooling + linear head: out[g] += dot(y[node], Ww); out pre-set to Wb.
// One wave per node; shfl_xor reduction over 32 lanes.
// ---------------------------------------------------------------------------
__global__ __launch_bounds__(256) void pool_kernel(
    const float* __restrict__ y, const long long* __restrict__ bidx,
    const float* __restrict__ Ww, float* __restrict__ out) {
  const int lane = threadIdx.x & 31;
  const int node = blockIdx.x * 8 + (threadIdx.x >> 5);
  const float* yr = y + (size_t)node * HID;
  float p = yr[lane] * Ww[lane] + yr[lane + 32] * Ww[lane + 32];
#pragma unroll
  for (int off = 16; off > 0; off >>= 1) p += __shfl_xor(p, off, 32);
  if (lane == 0) atomAddF(out + bidx[node], p);
}

extern "C" void kernel_launch(void* const* d_in, const int* in_sizes, int n_in,
                              void* d_out, int out_size, void* d_ws, size_t ws_size,
                              hipStream_t stream) {
  const float*     H    = (const float*)d_in[0];
  const float*     Xe   = (const float*)d_in[1];
  const long long* idXe = (const long long*)d_in[2];   // int64 [2, NE]
  const long long* bidx = (const long long*)d_in[3];   // int64 [NN]
  const float*     Mw0  = (const float*)d_in[4];
  const float*     Mb0  = (const float*)d_in[5];
  const float*     Uw0  = (const float*)d_in[6];
  const float*     Ub0  = (const float*)d_in[7];
  const float*     Mwh  = (const float*)d_in[8];       // [3, 80, 64]
  const float*     Mbh  = (const float*)d_in[9];       // [3, 64]
  const float*     Uwh  = (const float*)d_in[10];      // [3, 128, 64]
  const float*     Ubh  = (const float*)d_in[11];      // [3, 64]
  const float*     Ww   = (const float*)d_in[12];
  const float*     Wb   = (const float*)d_in[13];
  float* out = (float*)d_out;

  float* agg = (float*)d_ws;                 // [NN, 64]
  float* hA  = agg + (size_t)NN * HID;       // [NN, 64]
  float* hB  = hA  + (size_t)NN * HID;       // [NN, 64]
  const long long* src = idXe;
  const long long* dst = idXe + NE;

  const int nAgg = NN * HID;
  const dim3 blk(256);
  const int gZero = (nAgg + 255) / 256;   // 20000
  const int gMsg  = NE / (16 * 8);        // 10000 (exact)
  const int gUpd  = NN / (16 * 8);        // 625   (exact)
  const int gPool = NN / 8;               // 10000 (exact)

  // Layer 0: M (48->64), U (96->64)
  zero_f<<<gZero, blk, 0, stream>>>(agg, nAgg);
  msg_kernel<ND><<<gMsg, blk, 0, stream>>>(H, Xe, src, dst, Mw0, Mb0, agg);
  upd_kernel<ND><<<gUpd, blk, 0, stream>>>(H, agg, Uw0, Ub0, hA);

  // Hidden layers: M (80->64), U (128->64)
  float* cur = hA;
  float* nxt = hB;
  for (int i = 0; i < 3; ++i) {
    zero_f<<<gZero, blk, 0, stream>>>(agg, nAgg);
    msg_kernel<HID><<<gMsg, blk, 0, stream>>>(cur, Xe, src, dst,
                                              Mwh + (size_t)i * (HID + ED) * HID,
                                              Mbh + (size_t)i * HID, agg);
    upd_kernel<HID><<<gUpd, blk, 0, stream>>>(cur, agg,
                                              Uwh + (size_t)i * (2 * HID) * HID,
                                              Ubh + (size_t)i * HID, nxt);
    float* t = cur; cur = nxt; nxt = t;
  }

  // Pool + head
  init_out_k<<<1, blk, 0, stream>>>(out, Wb);
  pool_kernel<<<gPool, blk, 0, stream>>>(cur, bidx, Ww, out);
}